// GAT_Modeli_7121055777209
// MI455X (gfx1250) — compile-verified
//
#include <hip/hip_runtime.h>

#define N_NODES   50000
#define N_EDGES   1600000
#define E_TOT     1650000
#define HID       64
#define HEADS     4
#define HC1       256        // HEADS*HID
#define IN_CH     7
#define OUT_CH    6
#define NEG_SLOPE 0.2f

typedef __attribute__((ext_vector_type(2))) float v2f;
typedef __attribute__((ext_vector_type(8))) float v8f;

// ---- float <-> order-preserving unsigned (for atomicMax on floats) ----
static __device__ __forceinline__ unsigned fenc(float f) {
  unsigned u = __float_as_uint(f);
  return (u & 0x80000000u) ? ~u : (u | 0x80000000u);
}
static __device__ __forceinline__ float fdec(unsigned u) {
  unsigned b = (u & 0x80000000u) ? (u & 0x7fffffffu) : ~u;
  return __uint_as_float(b);
}
static __device__ __forceinline__ float lrelu(float x) {
  return x > 0.f ? x : NEG_SLOPE * x;
}
static __device__ __forceinline__ void edge_sd(const long long* __restrict__ ei,
                                               int e, int& s, int& d) {
  if (e < N_EDGES) { s = (int)ei[e]; d = (int)ei[N_EDGES + e]; }
  else             { s = d = e - N_EDGES; }             // self-loops appended
}

// ================= GEMM1: h1[50000,256] = x[50000,7] @ W1[7,256] ============
// one wave per 16x16 output tile; 3125 * 16 tiles; K=7 -> two 16x16x4 WMMAs
__global__ __launch_bounds__(256) void gemm1_wmma(const float* __restrict__ x,
                                                  const float* __restrict__ W1,
                                                  float* __restrict__ h1) {
  int wid  = (blockIdx.x * blockDim.x + threadIdx.x) >> 5;   // wave id (uniform)
  int lane = threadIdx.x & 31;
  int tm = wid >> 4, tn = wid & 15;                          // 3125 x 16 tiles
  int row  = tm * 16 + (lane & 15);                          // A: lane&15 = M
  int ncol = tn * 16 + (lane & 15);                          // B: lane&15 = N
  int kh   = (lane < 16) ? 0 : 2;                            // K half per lane group
  v8f c = {};
  // K = 0..3
  v2f a0, b0;
  a0.x = x[row * IN_CH + kh];
  a0.y = x[row * IN_CH + kh + 1];
  b0.x = W1[(kh)     * HC1 + ncol];
  b0.y = W1[(kh + 1) * HC1 + ncol];
  c = __builtin_amdgcn_wmma_f32_16x16x4_f32(false, a0, false, b0, (short)0, c, false, false);
  // K = 4..6 (K=7 zero-padded)
  int k2 = 4 + kh;
  v2f a1, b1;
  a1.x = x[row * IN_CH + k2];
  a1.y = (k2 + 1 < IN_CH) ? x[row * IN_CH + k2 + 1] : 0.f;
  b1.x = W1[k2 * HC1 + ncol];
  b1.y = (k2 + 1 < IN_CH) ? W1[(k2 + 1) * HC1 + ncol] : 0.f;
  c = __builtin_amdgcn_wmma_f32_16x16x4_f32(false, a1, false, b1, (short)0, c, false, false);
  // store C: lanes 0-15 -> M = r, lanes 16-31 -> M = 8 + r ; N = lane&15
  int mbase = tm * 16 + ((lane >= 16) ? 8 : 0);
#pragma unroll
  for (int r = 0; r < 8; ++r)
    h1[(size_t)(mbase + r) * HC1 + ncol] = c[r];
}

// ===== GEMM2: h2[50000,6] = hmid[50000,256] @ W2p[256,16] (cols 6..15 = 0) ===
__global__ __launch_bounds__(256) void gemm2_wmma(const float* __restrict__ hm,
                                                  const float* __restrict__ W2p,
                                                  float* __restrict__ h2) {
  int wid  = (blockIdx.x * blockDim.x + threadIdx.x) >> 5;
  int lane = threadIdx.x & 31;
  if (wid >= N_NODES / 16) return;                           // wave-uniform
  int row  = wid * 16 + (lane & 15);
  int ncol = lane & 15;
  int kh   = (lane < 16) ? 0 : 2;
  v8f c = {};
  for (int k = 0; k < HC1; k += 4) {
    v2f a, b;
    a.x = hm[(size_t)row * HC1 + k + kh];
    a.y = hm[(size_t)row * HC1 + k + kh + 1];
    b.x = W2p[(k + kh)     * 16 + ncol];
    b.y = W2p[(k + kh + 1) * 16 + ncol];
    c = __builtin_amdgcn_wmma_f32_16x16x4_f32(false, a, false, b, (short)0, c, false, false);
  }
  int mbase = wid * 16 + ((lane >= 16) ? 8 : 0);
  if (ncol < OUT_CH) {
#pragma unroll
    for (int r = 0; r < 8; ++r)
      h2[(size_t)(mbase + r) * OUT_CH + ncol] = c[r];
  }
}

__global__ void pad_w2(const float* __restrict__ W2, float* __restrict__ W2p) {
  int t = blockIdx.x * blockDim.x + threadIdx.x;             // 256*16
  if (t >= HC1 * 16) return;
  int k = t >> 4, n = t & 15;
  W2p[t] = (n < OUT_CH) ? W2[k * OUT_CH + n] : 0.f;
}

// ================= attention coefficients ===================================
__global__ void attn_coef1(const float* __restrict__ h1, const float* __restrict__ asr,
                           const float* __restrict__ ads, float* __restrict__ als,
                           float* __restrict__ ald) {
  int t = blockIdx.x * blockDim.x + threadIdx.x;             // node*H + h
  if (t >= N_NODES * HEADS) return;
  int n = t >> 2, h = t & 3;
  const float* hp = h1 + (size_t)n * HC1 + h * HID;
  const float* as = asr + h * HID;
  const float* ad = ads + h * HID;
  float s = 0.f, d = 0.f;
#pragma unroll 8
  for (int cc = 0; cc < HID; ++cc) { float v = hp[cc]; s += v * as[cc]; d += v * ad[cc]; }
  als[t] = s; ald[t] = d;
}

__global__ void attn_coef2(const float* __restrict__ h2, const float* __restrict__ asr,
                           const float* __restrict__ ads, float* __restrict__ als,
                           float* __restrict__ ald) {
  int n = blockIdx.x * blockDim.x + threadIdx.x;
  if (n >= N_NODES) return;
  float s = 0.f, d = 0.f;
#pragma unroll
  for (int c = 0; c < OUT_CH; ++c) {
    float v = h2[(size_t)n * OUT_CH + c]; s += v * asr[c]; d += v * ads[c];
  }
  als[n] = s; ald[n] = d;
}

// ================= edge softmax passes ======================================
template <int H>
__global__ void edge_max_k(const long long* __restrict__ ei, const float* __restrict__ als,
                           const float* __restrict__ ald, unsigned* __restrict__ m) {
  int t = blockIdx.x * blockDim.x + threadIdx.x;
  if (t >= E_TOT * H) return;
  int e = t / H, h = t % H;
  int s, d; edge_sd(ei, e, s, d);
  float ev = lrelu(als[(size_t)s * H + h] + ald[(size_t)d * H + h]);
  atomicMax(&m[(size_t)d * H + h], fenc(ev));
}

template <int H>
__global__ void edge_expsum_k(const long long* __restrict__ ei, const float* __restrict__ als,
                              const float* __restrict__ ald, const unsigned* __restrict__ m,
                              float* __restrict__ denom, float* __restrict__ ex) {
  int t = blockIdx.x * blockDim.x + threadIdx.x;
  if (t >= E_TOT * H) return;
  int e = t / H, h = t % H;
  int s, d; edge_sd(ei, e, s, d);
  float ev = lrelu(als[(size_t)s * H + h] + ald[(size_t)d * H + h]);
  float x = expf(ev - fdec(m[(size_t)d * H + h]));
  ex[t] = x;                                   // per-edge unnormalized weight
  atomicAdd(&denom[(size_t)d * H + h], x);
}

// layer1 aggregate: thread = (edge, 4-channel group); float4 gather of h1[src]
__global__ void edge_aggr1(const long long* __restrict__ ei, const float* __restrict__ h1,
                           const float* __restrict__ ex, const float* __restrict__ denom,
                           float* __restrict__ agg) {
  int t = blockIdx.x * blockDim.x + threadIdx.x;             // e*64 + g
  if (t >= E_TOT * 64) return;
  int e = t >> 6, g = t & 63;
  int s, d; edge_sd(ei, e, s, d);
  int h = g >> 4;
  float a = ex[(size_t)e * HEADS + h] / denom[(size_t)d * HEADS + h];
  const float4 v = *(const float4*)(h1 + (size_t)s * HC1 + g * 4);
  float* o = agg + (size_t)d * HC1 + g * 4;
  atomicAdd(o + 0, v.x * a); atomicAdd(o + 1, v.y * a);
  atomicAdd(o + 2, v.z * a); atomicAdd(o + 3, v.w * a);
}

__global__ void edge_aggr2(const long long* __restrict__ ei, const float* __restrict__ h2,
                           const float* __restrict__ ex, const float* __restrict__ denom,
                           float* __restrict__ agg) {
  int t = blockIdx.x * blockDim.x + threadIdx.x;             // e*8 + c
  if (t >= E_TOT * 8) return;
  int e = t >> 3, c = t & 7;
  if (c >= OUT_CH) return;
  int s, d; edge_sd(ei, e, s, d);
  float a = ex[e] / denom[d];
  atomicAdd(&agg[(size_t)d * OUT_CH + c], h2[(size_t)s * OUT_CH + c] * a);
}

// ================= elementwise ==============================================
__global__ void bias_elu(float* __restrict__ agg, const float* __restrict__ b1) {
  int t = blockIdx.x * blockDim.x + threadIdx.x;
  if (t >= N_NODES * HC1) return;
  float v = agg[t] + b1[t & (HC1 - 1)];
  agg[t] = v > 0.f ? v : (expf(v) - 1.f);
}

__global__ void final_lsm(const float* __restrict__ agg2, const float* __restrict__ b2,
                          float* __restrict__ out) {
  int n = blockIdx.x * blockDim.x + threadIdx.x;
  if (n >= N_NODES) return;
  float z[OUT_CH];
  float mx = -3.4e38f;
#pragma unroll
  for (int c = 0; c < OUT_CH; ++c) {
    z[c] = agg2[(size_t)n * OUT_CH + c] + b2[c];
    mx = fmaxf(mx, z[c]);
  }
  float sm = 0.f;
#pragma unroll
  for (int c = 0; c < OUT_CH; ++c) sm += expf(z[c] - mx);
  float ls = logf(sm);
#pragma unroll
  for (int c = 0; c < OUT_CH; ++c) out[(size_t)n * OUT_CH + c] = z[c] - mx - ls;
}

// ================= launch ===================================================
static inline int cdiv(long long a, int b) { return (int)((a + b - 1) / b); }

extern "C" void kernel_launch(void* const* d_in, const int* in_sizes, int n_in,
                              void* d_out, int out_size, void* d_ws, size_t ws_size,
                              hipStream_t stream) {
  (void)in_sizes; (void)n_in; (void)out_size; (void)ws_size;
  const float*     x    = (const float*)d_in[0];
  const long long* ei   = (const long long*)d_in[1];     // int64 [2, E]
  const float*     W1   = (const float*)d_in[2];
  const float*     as1  = (const float*)d_in[3];
  const float*     ad1  = (const float*)d_in[4];
  const float*     b1   = (const float*)d_in[5];
  const float*     W2   = (const float*)d_in[6];
  const float*     as2  = (const float*)d_in[7];
  const float*     ad2  = (const float*)d_in[8];
  const float*     b2   = (const float*)d_in[9];
  float*           out  = (float*)d_out;

  float* ws = (float*)d_ws;
  size_t o = 0;
  float*    h1    = ws + o; o += (size_t)N_NODES * HC1;      // 12.8M
  float*    agg1  = ws + o; o += (size_t)N_NODES * HC1;      // 12.8M (becomes hmid)
  float*    als1  = ws + o; o += (size_t)N_NODES * HEADS;
  float*    ald1  = ws + o; o += (size_t)N_NODES * HEADS;
  unsigned* m1    = (unsigned*)(ws + o); o += (size_t)N_NODES * HEADS;
  float*    den1  = ws + o; o += (size_t)N_NODES * HEADS;
  float*    ex    = ws + o; o += (size_t)E_TOT * HEADS;      // 6.6M (reused L2)
  float*    h2    = ws + o; o += (size_t)N_NODES * OUT_CH;
  float*    als2  = ws + o; o += (size_t)N_NODES;
  float*    ald2  = ws + o; o += (size_t)N_NODES;
  unsigned* m2    = (unsigned*)(ws + o); o += (size_t)N_NODES;
  float*    den2  = ws + o; o += (size_t)N_NODES;
  float*    agg2  = ws + o; o += (size_t)N_NODES * OUT_CH;
  float*    W2p   = ws + o; o += (size_t)HC1 * 16;

  // zero accumulators (fenc: 0u == "less than every float")
  hipMemsetAsync(m1,   0, sizeof(unsigned) * N_NODES * HEADS, stream);
  hipMemsetAsync(den1, 0, sizeof(float) * N_NODES * HEADS, stream);
  hipMemsetAsync(agg1, 0, sizeof(float) * (size_t)N_NODES * HC1, stream);
  hipMemsetAsync(m2,   0, sizeof(unsigned) * N_NODES, stream);
  hipMemsetAsync(den2, 0, sizeof(float) * N_NODES, stream);
  hipMemsetAsync(agg2, 0, sizeof(float) * N_NODES * OUT_CH, stream);

  const int B = 256;

  // ---- layer 1 ----
  gemm1_wmma<<<(N_NODES / 16) * 16 * 32 / B, B, 0, stream>>>(x, W1, h1);
  attn_coef1<<<cdiv(N_NODES * HEADS, B), B, 0, stream>>>(h1, as1, ad1, als1, ald1);
  edge_max_k<HEADS><<<cdiv((long long)E_TOT * HEADS, B), B, 0, stream>>>(ei, als1, ald1, m1);
  edge_expsum_k<HEADS><<<cdiv((long long)E_TOT * HEADS, B), B, 0, stream>>>(ei, als1, ald1, m1, den1, ex);
  edge_aggr1<<<cdiv((long long)E_TOT * 64, B), B, 0, stream>>>(ei, h1, ex, den1, agg1);
  bias_elu<<<cdiv((long long)N_NODES * HC1, B), B, 0, stream>>>(agg1, b1);

  // ---- layer 2 ----
  pad_w2<<<cdiv(HC1 * 16, B), B, 0, stream>>>(W2, W2p);
  gemm2_wmma<<<cdiv((long long)(N_NODES / 16) * 32, B), B, 0, stream>>>(agg1, W2p, h2);
  attn_coef2<<<cdiv(N_NODES, B), B, 0, stream>>>(h2, as2, ad2, als2, ald2);
  edge_max_k<1><<<cdiv((long long)E_TOT, B), B, 0, stream>>>(ei, als2, ald2, m2);
  edge_expsum_k<1><<<cdiv((long long)E_TOT, B), B, 0, stream>>>(ei, als2, ald2, m2, den2, ex);
  edge_aggr2<<<cdiv((long long)E_TOT * 8, B), B, 0, stream>>>(ei, h2, ex, den2, agg2);
  final_lsm<<<cdiv(N_NODES, B), B, 0, stream>>>(agg2, b2, out);
}